// CubeDiagonalAttention_67534065762984
// MI455X (gfx1250) — compile-verified
//
#include <hip/hip_runtime.h>
#include <hip/hip_bf16.h>

typedef __attribute__((ext_vector_type(2))) float v2f;
typedef __attribute__((ext_vector_type(4))) float v4f;
typedef __attribute__((ext_vector_type(8))) float v8f;

#define D_MODEL 1024
#define N_TOK   4096
#define N_BATCH 4
#define N_ROWS  (N_BATCH * N_TOK)   // 16384

// ---------------------------------------------------------------------------
// Kernel 1: z = x . W^T via V_WMMA_F32_16X16X4_F32, then 3-bit sign code/row.
//
// A = W tile (16x4, rows>=3 zeroed), B = x^T tile (4x16).
// D(16x16 f32): lane L<16 owns z[k][row=rowBase+L] for k=0..7 in c[0..7],
// so the 3-bit sign code (k=0..2) is lane-local.
//
// W rows 3..15 are zero-padded WITHOUT exec-mask branches: the pointer is
// clamped to a valid row and the loaded value is zeroed via a value select
// (v_cndmask), keeping the loop body straight-line.
// ---------------------------------------------------------------------------
__global__ __launch_bounds__(32)
void cda_proj_sign_kernel(const float* __restrict__ x,
                          const float* __restrict__ W,
                          unsigned char* __restrict__ codes) {
    const int lane = threadIdx.x;          // 0..31 (wave32)
    const int half = lane >> 4;            // 0: K0/K1, 1: K2/K3
    const int m    = lane & 15;            // A-row / B-col index
    const long long rowBase = (long long)blockIdx.x * 16;

    const bool  wvalid = (m < 3);
    const int   wm     = wvalid ? m : 0;   // clamp -> always-safe address
    const float* __restrict__ xrow = x + (rowBase + m) * D_MODEL;
    const float* __restrict__ wrow = W + wm * D_MODEL;

    v8f c = {};
#pragma unroll 4
    for (int s = 0; s < D_MODEL / 4; ++s) {
        const int d = 4 * s + 2 * half;

        // A fragment: one b64 load, then value-select to zero invalid rows.
        v2f aw = *(const v2f*)(wrow + d);
        v2f a;
        a.x = wvalid ? aw.x : 0.0f;
        a.y = wvalid ? aw.y : 0.0f;

        // B fragment: x^T[d][m] = x[rowBase+m][d], one b64 load.
        v2f b = *(const v2f*)(xrow + d);

        // (neg_a, A, neg_b, B, c_mod, C, reuse_a, reuse_b)
        c = __builtin_amdgcn_wmma_f32_16x16x4_f32(
                false, a, false, b, (short)0, c, false, false);
    }

    if (lane < 16) {
        unsigned code = (c[0] > 0.0f ? 1u : 0u)
                      | (c[1] > 0.0f ? 2u : 0u)
                      | (c[2] > 0.0f ? 4u : 0u);
        codes[rowBase + lane] = (unsigned char)code;
    }
}

// ---------------------------------------------------------------------------
// Kernel 2: out[b][i][j] = dw[popcount(code[b][i] ^ code[b][j])]
// One thread -> 4 consecutive columns (one NT float4 store). Pure store-BW;
// 268 MB stream exceeds the 192 MB L2, so stores are non-temporal.
// ---------------------------------------------------------------------------
__global__ __launch_bounds__(256)
void cda_bias_kernel(const unsigned char* __restrict__ codes,
                     const float* __restrict__ dw,
                     float* __restrict__ out) {
    const long long t   = (long long)blockIdx.x * blockDim.x + threadIdx.x; // 0..16M-1
    const int       j4  = (int)(t & (N_TOK / 4 - 1));     // column group of 4
    const long long row = t >> 10;                        // b*4096 + i
    const int       b   = (int)(row >> 12);

    const unsigned ci = codes[row];
    const unsigned* __restrict__ codes32 =
        (const unsigned*)(codes + ((long long)b << 12));
    const unsigned cj4 = codes32[j4];                     // 4 packed col codes

    float tab[4] = {dw[0], dw[1], dw[2], dw[3]};

    v4f o;
#pragma unroll
    for (int e = 0; e < 4; ++e) {
        const unsigned cj = (cj4 >> (8 * e)) & 0xFFu;
        o[e] = tab[__popc((ci ^ cj) & 7u)];
    }

    __builtin_nontemporal_store(o, (v4f*)(out + (t << 2)));
}

extern "C" void kernel_launch(void* const* d_in, const int* in_sizes, int n_in,
                              void* d_out, int out_size, void* d_ws, size_t ws_size,
                              hipStream_t stream) {
    const float* x  = (const float*)d_in[0];   // [4, 4096, 1024] f32
    const float* W  = (const float*)d_in[1];   // [3, 1024] f32
    const float* dw = (const float*)d_in[2];   // [4] f32
    float* out = (float*)d_out;                // [4, 4096, 4096] f32

    unsigned char* codes = (unsigned char*)d_ws;   // 16384 bytes

    // Stage 1: 16384 rows / 16 rows-per-wave = 1024 single-wave blocks.
    cda_proj_sign_kernel<<<N_ROWS / 16, 32, 0, stream>>>(x, W, codes);

    // Stage 2: 4*4096*4096 floats / 4 per thread = 16,777,216 threads.
    const long long n_threads = (long long)N_BATCH * N_TOK * N_TOK / 4;
    const int block = 256;
    const int grid  = (int)(n_threads / block);    // 65536
    cda_bias_kernel<<<grid, block, 0, stream>>>(codes, dw, out);
}